// Track_Loss_18056042512996
// MI455X (gfx1250) — compile-verified
//
#include <hip/hip_runtime.h>
#include <math.h>

#define ALPHA_C 0.25f
#define T0_C    0.5f
#define T1_C    0.4f
#define EPS_C   1e-7f

typedef __attribute__((ext_vector_type(2))) float v2f;
typedef __attribute__((ext_vector_type(8))) float v8f;

// ---------------------------------------------------------------------------
// Exact f32 wave(32) sum on the matrix unit.
// A (16x4): lane L<16 holds A[L][0]=v, A[L][1]=0 ; lane L>=16 holds A[L-16][2]=v,
// A[L-16][3]=0.  B (4x16) = ones.  D[m][n] = v[m] + v[m+16] for every n.
// Summing the 8 D VGPRs gives half-row sums; one xor-16 shuffle completes it.
// Requires EXEC all-ones (call only from convergent code, full blocks).
// ---------------------------------------------------------------------------
__device__ __forceinline__ float wave_sum32(float v) {
    v2f a = {v, 0.0f};
    v2f b = {1.0f, 1.0f};
    v8f c = {};
    c = __builtin_amdgcn_wmma_f32_16x16x4_f32(false, a, false, b,
                                              (short)0, c, false, false);
    float s = ((c[0] + c[1]) + (c[2] + c[3])) + ((c[4] + c[5]) + (c[6] + c[7]));
    s += __shfl_xor(s, 16, 32);   // combine m=0..7 half with m=8..15 half
    return s;                     // full wave sum in every lane
}

// stable log(exp(a)+exp(b))
__device__ __forceinline__ float lse2(float a, float b) {
    return fmaxf(a, b) + log1pf(__expf(-fabsf(a - b)));
}

// ---------------------------------------------------------------------------
// RPN: focal loss partials + ltrb-IoU partials + positive count.
// Grid-stride, uniform trip count (total % (grid*block) == 0 for our launch).
// ---------------------------------------------------------------------------
__global__ void rpn_kernel(const float* __restrict__ cl,
                           const float* __restrict__ re,
                           const float* __restrict__ gr,
                           const float* __restrict__ gt,
                           float* __restrict__ pf, int total) {
    const float2* cl2 = (const float2*)cl;
    const float4* re4 = (const float4*)re;
    const float4* gr4 = (const float4*)gr;

    float s_f = 0.0f, s_i = 0.0f, s_p = 0.0f;
    for (int i = blockIdx.x * blockDim.x + threadIdx.x; i < total;
         i += gridDim.x * blockDim.x) {
        float2 c = cl2[i];
        float  g = gt[i];
        bool   fg = (g > 0.0f);                 // tgt = 1-gt ; fg <=> tgt==0

        float l   = lse2(c.x, c.y);
        float lpt = (fg ? c.x : c.y) - l;       // logp[tgt]
        float at  = fg ? ALPHA_C : (1.0f - ALPHA_C);
        float pt  = __expf(lpt);
        float om  = 1.0f - pt;
        s_f += -(om * om) * at * lpt;           // GAMMA == 2

        float4 r = re4[i], q = gr4[i];
        float m0 = fminf(r.x, q.x), m1 = fminf(r.y, q.y);
        float m2 = fminf(r.z, q.z), m3 = fminf(r.w, q.w);
        float inter = (m0 + m2) * (m1 + m3);
        float ag = (q.x + q.z) * (q.y + q.w);
        float ar = (r.x + r.z) * (r.y + r.w);
        float un = ag + ar - inter + EPS_C;
        float iou = (inter + 1.0f) / (un + 1.0f);
        float pm = fg ? 1.0f : 0.0f;
        s_i += (1.0f - iou) * pm;
        s_p += pm;
    }

    __shared__ float lds[8][3];
    float w0 = wave_sum32(s_f);
    float w1 = wave_sum32(s_i);
    float w2 = wave_sum32(s_p);
    int wid = threadIdx.x >> 5, lane = threadIdx.x & 31;
    if (lane == 0) { lds[wid][0] = w0; lds[wid][1] = w1; lds[wid][2] = w2; }
    __syncthreads();
    if (threadIdx.x == 0) {
        float a = 0.f, b = 0.f, c = 0.f;
        int nw = (int)(blockDim.x >> 5);
        for (int i = 0; i < nw; ++i) { a += lds[i][0]; b += lds[i][1]; c += lds[i][2]; }
        pf[blockIdx.x]                 = a;
        pf[gridDim.x + blockIdx.x]     = b;
        pf[2 * gridDim.x + blockIdx.x] = c;
    }
}

// ---------------------------------------------------------------------------
// RCNN: one block per batch image. 2048 proposals, 8 per thread.
// ---------------------------------------------------------------------------
__global__ void rcnn_kernel(const float* __restrict__ cf,
                            const float* __restrict__ op,
                            const float* __restrict__ bb,
                            const float* __restrict__ br,
                            const float* __restrict__ gb,
                            float* __restrict__ outLoss,
                            float* __restrict__ outPcnt,
                            int N, int K) {
    const int b = blockIdx.x;
    const float g0 = gb[b * 4 + 0], g1 = gb[b * 4 + 1];
    const float g2 = gb[b * 4 + 2], g3 = gb[b * 4 + 3];
    const float area_a = (g2 - g0) * (g3 - g1);

    const float4* br4 = (const float4*)br + (size_t)b * N;
    const float4* bb4 = (const float4*)bb + (size_t)b * N;
    const float*  opb = op + (size_t)b * N;
    const int     cq  = K / 2;                       // float4s per proposal (=8)
    const float4* cf4 = (const float4*)cf + (size_t)b * N * cq;

    float s_pos = 0.f, s_neg = 0.f, s_pp = 0.f, s_pn = 0.f;
    float s_ng = 0.f, s_op = 0.f, s_bb = 0.f;

    for (int n = threadIdx.x; n < N; n += blockDim.x) {
        // jaccard(gb, br, EPS)
        float4 rb = br4[n];
        float wx = fmaxf(fminf(g2, rb.z) - fmaxf(g0, rb.x), 0.f);
        float wy = fmaxf(fminf(g3, rb.w) - fmaxf(g1, rb.y), 0.f);
        float inter  = wx * wy;
        float area_b = (rb.z - rb.x) * (rb.w - rb.y);
        float iou    = (inter + EPS_C) / (area_a + area_b - inter + EPS_C);
        float pos = (iou >= T0_C) ? 1.f : 0.f;
        float neg = (iou <  T1_C) ? 1.f : 0.f;
        s_pos += pos; s_neg += neg;

        const float4* cfn = cf4 + (size_t)n * cq;
        __builtin_prefetch(cfn + (size_t)blockDim.x * cq, 0, 1); // next tile -> global_prefetch_b8
        float ce = 0.f;
        {
            float4 v = cfn[0];                       // k=0 (x,y), k=1 (z,w)
            float l0 = lse2(v.x, v.y);
            s_pp += (l0 - v.x) * pos;                // -logp[:, :, 0, 0]
            s_pn += (l0 - v.y) * neg;                // -logp[:, :, 0, 1]
            ce   += (lse2(v.z, v.w) - v.w);          // k=1 contribution
        }
        for (int j = 1; j < cq; ++j) {
            float4 v = cfn[j];
            ce += (lse2(v.x, v.y) - v.y);
            ce += (lse2(v.z, v.w) - v.w);
        }
        s_ng += ce * pos;

        float x   = opb[n];
        float bce = fmaxf(x, 0.f) - x * iou + log1pf(__expf(-fabsf(x)));
        s_op += bce * pos;

        // jaccard(gb, bb, eps=1.0)
        float4 qb = bb4[n];
        float wx2 = fmaxf(fminf(g2, qb.z) - fmaxf(g0, qb.x), 0.f);
        float wy2 = fmaxf(fminf(g3, qb.w) - fmaxf(g1, qb.y), 0.f);
        float inter2  = wx2 * wy2;
        float area_b2 = (qb.z - qb.x) * (qb.w - qb.y);
        float iou2 = (inter2 + 1.0f) / (area_a + area_b2 - inter2 + 1.0f);
        s_bb += (1.f - iou2) * pos;
    }

    __shared__ float lds[8][7];
    float w[7] = {s_pos, s_neg, s_pp, s_pn, s_ng, s_op, s_bb};
#pragma unroll
    for (int j = 0; j < 7; ++j) w[j] = wave_sum32(w[j]);
    int wid = threadIdx.x >> 5, lane = threadIdx.x & 31;
    if (lane == 0)
        for (int j = 0; j < 7; ++j) lds[wid][j] = w[j];
    __syncthreads();
    if (threadIdx.x == 0) {
        float t[7] = {0, 0, 0, 0, 0, 0, 0};
        int nw = (int)(blockDim.x >> 5);
        for (int i = 0; i < nw; ++i)
            for (int j = 0; j < 7; ++j) t[j] += lds[i][j];
        float pcnt = t[0], ncnt = t[1];
        bool  pp = (pcnt > 0.f);
        float l_pp = pp ? t[2] / fmaxf(pcnt, 1.f) : 0.f;
        float l_pn = (ncnt > 0.f) ? t[3] / fmaxf(ncnt, 1.f) : 0.f;
        float l_ng = pp ? t[4] / fmaxf(pcnt * (float)(K - 1), 1.f) : 0.f;
        float l_op = pp ? t[5] / fmaxf(pcnt, 1.f) : 0.f;
        float l_bb = pp ? t[6] / fmaxf(pcnt, 1.f) : 0.f;
        outLoss[blockIdx.x] = pp ? (l_pp + l_pn + l_ng + l_bb + l_op) : 0.f;
        outPcnt[blockIdx.x] = pcnt;
    }
}

// ---------------------------------------------------------------------------
// Finalize: fold 1024 RPN partials + 128 batch results into the 5 outputs.
// ---------------------------------------------------------------------------
__global__ void finalize_kernel(const float* __restrict__ pf, int nblk,
                                const float* __restrict__ lossi,
                                const float* __restrict__ pcnt, int B,
                                float* __restrict__ out, float inv_total) {
    float a = 0.f, bsum = 0.f, c = 0.f, rl = 0.f, pc = 0.f;
    for (int i = threadIdx.x; i < nblk; i += blockDim.x) {
        a    += pf[i];
        bsum += pf[nblk + i];
        c    += pf[2 * nblk + i];
    }
    for (int i = threadIdx.x; i < B; i += blockDim.x) {
        rl += lossi[i];
        pc += pcnt[i];
    }
    __shared__ float lds[8][5];
    float w[5] = {a, bsum, c, rl, pc};
#pragma unroll
    for (int j = 0; j < 5; ++j) w[j] = wave_sum32(w[j]);
    int wid = threadIdx.x >> 5, lane = threadIdx.x & 31;
    if (lane == 0)
        for (int j = 0; j < 5; ++j) lds[wid][j] = w[j];
    __syncthreads();
    if (threadIdx.x == 0) {
        float t[5] = {0, 0, 0, 0, 0};
        int nw = (int)(blockDim.x >> 5);
        for (int i = 0; i < nw; ++i)
            for (int j = 0; j < 5; ++j) t[j] += lds[i][j];
        float rpn0 = t[0] * inv_total;
        float npos = t[2];
        float rpn1 = (npos > 0.f) ? t[1] / fmaxf(npos, 1.f) : 0.f;
        float rcnn = t[3] / (float)B;
        float total_pos = t[4];
        out[0] = rpn0 + rpn1 + 1.0f * rcnn;     // LAM = 1
        out[1] = rpn0;
        out[2] = rpn1;
        out[3] = rcnn;
        out[4] = (float)(long long)total_pos;   // astype(int32)
    }
}

extern "C" void kernel_launch(void* const* d_in, const int* in_sizes, int n_in,
                              void* d_out, int out_size, void* d_ws, size_t ws_size,
                              hipStream_t stream) {
    (void)n_in; (void)out_size; (void)ws_size;
    const float* cl = (const float*)d_in[0];
    const float* re = (const float*)d_in[1];
    const float* cf = (const float*)d_in[2];
    const float* op = (const float*)d_in[3];
    const float* bb = (const float*)d_in[4];
    const float* br = (const float*)d_in[5];
    const float* gb = (const float*)d_in[6];
    const float* gt = (const float*)d_in[7];
    const float* gr = (const float*)d_in[8];

    const int B     = in_sizes[6] / 4;            // gb: (B,4)
    const int total = in_sizes[7];                // gt: (B,HW)
    const int N     = in_sizes[3] / B;            // op: (B,N,1)
    const int K     = in_sizes[2] / (B * N * 2);  // cf: (B,N,K,2)

    float* W     = (float*)d_ws;
    const int NB = 1024;                          // total=2M -> exactly 8 elem/thread
    float* pf    = W;                             // [3*NB]
    float* lossi = W + 3 * NB;                    // [B]
    float* pcn   = W + 3 * NB + B;                // [B]

    rpn_kernel<<<NB, 256, 0, stream>>>(cl, re, gr, gt, pf, total);
    rcnn_kernel<<<B, 256, 0, stream>>>(cf, op, bb, br, gb, lossi, pcn, N, K);
    finalize_kernel<<<1, 256, 0, stream>>>(pf, NB, lossi, pcn, B,
                                           (float*)d_out, 1.0f / (float)total);
}